// CGMMLayer_74363063763466
// MI455X (gfx1250) — compile-verified
//
#include <hip/hip_runtime.h>
#include <math.h>

// CGMM layer for MI455X (gfx1250, wave32).
// Phases: softmax(Q,B) -> edge scatter-mean (L2-resident atomics) ->
//         per-node 16x16x16 fp32 matmul via V_WMMA_F32_16X16X4_F32.
//
// Workspace layout (floats): Qs[2048] | Bs[4096] | cnt[N] | summed[N*128]
//   ~25.9 MB total.

typedef __attribute__((ext_vector_type(2))) float v2f;
typedef __attribute__((ext_vector_type(8))) float v8f;

#define C_DIM 16
#define M_DIM 32
#define G_DIM 8
#define ROW   (C_DIM * G_DIM)   // 128 floats per node row

// XOR-butterfly swap within 32 lanes via ds_swizzle_b32 (group-of-32 mode):
// offset = (xor_mask << 10) | and_mask(0x1F). Single DS op, no bpermute index math.
__device__ __forceinline__ float swz_xor_add(float v, const int imm_off) {
    return v; // placeholder (specialized below via macro)
}
#define SWZ_ADD(v, XOFF) \
    ((v) + __int_as_float(__builtin_amdgcn_ds_swizzle(__float_as_int(v), XOFF)))

// ---------------- kernel: zero accumulators ----------------
__global__ void k_zero(float* __restrict__ p, long n) {
    long i = (long)blockIdx.x * blockDim.x + threadIdx.x;
    if (i < n) p[i] = 0.0f;
}

// ---------------- kernel: softmaxes of lambda_Q (axis=0) and lambda_B (axis=1)
__global__ void k_softmax(const float* __restrict__ lamQ,
                          const float* __restrict__ lamB,
                          float* __restrict__ Qs,
                          float* __restrict__ Bs) {
    int t = threadIdx.x;
    if (t < 128) {
        // Q[:, l, g] softmax over i (16 values)
        int l = t >> 3, g = t & 7;
        float mx = -1e30f;
        for (int i = 0; i < C_DIM; ++i)
            mx = fmaxf(mx, lamQ[(i * C_DIM + l) * G_DIM + g]);
        float s = 0.0f;
        for (int i = 0; i < C_DIM; ++i)
            s += expf(lamQ[(i * C_DIM + l) * G_DIM + g] - mx);
        float inv = 1.0f / s;
        for (int i = 0; i < C_DIM; ++i)
            Qs[(i * C_DIM + l) * G_DIM + g] =
                expf(lamQ[(i * C_DIM + l) * G_DIM + g] - mx) * inv;
    } else {
        // B[i, :, g] softmax over m (32 values)
        int p = t - 128;
        int i = p >> 3, g = p & 7;
        float mx = -1e30f;
        for (int m = 0; m < M_DIM; ++m)
            mx = fmaxf(mx, lamB[(i * M_DIM + m) * G_DIM + g]);
        float s = 0.0f;
        for (int m = 0; m < M_DIM; ++m)
            s += expf(lamB[(i * M_DIM + m) * G_DIM + g] - mx);
        float inv = 1.0f / s;
        for (int m = 0; m < M_DIM; ++m)
            Bs[(i * M_DIM + m) * G_DIM + g] =
                expf(lamB[(i * M_DIM + m) * G_DIM + g] - mx) * inv;
    }
}

// ---------------- kernel: scatter-sum of prev_h[src] rows onto summed[dst]
// one wave (32 lanes) per edge; float4 gather, 4 f32 atomics per lane.
// summed (25.6MB) and prev_h (25.6MB) are both L2-resident (192MB L2).
__global__ void k_scatter(const int* __restrict__ ei,
                          const float* __restrict__ prev_h,
                          float* __restrict__ summed,
                          float* __restrict__ cnt,
                          int E) {
    long t = (long)blockIdx.x * blockDim.x + threadIdx.x;
    int e = (int)(t >> 5);
    if (e >= E) return;
    int lane = (int)(t & 31);
    int dst = ei[e];        // edge_index[0, e]
    int src = ei[E + e];    // edge_index[1, e]

    const float4* sp = (const float4*)(prev_h + (long)src * ROW) + lane;
    float4 v = *sp;
    float* dp = summed + (long)dst * ROW + lane * 4;
    __hip_atomic_fetch_add(dp + 0, v.x, __ATOMIC_RELAXED, __HIP_MEMORY_SCOPE_AGENT);
    __hip_atomic_fetch_add(dp + 1, v.y, __ATOMIC_RELAXED, __HIP_MEMORY_SCOPE_AGENT);
    __hip_atomic_fetch_add(dp + 2, v.z, __ATOMIC_RELAXED, __HIP_MEMORY_SCOPE_AGENT);
    __hip_atomic_fetch_add(dp + 3, v.w, __ATOMIC_RELAXED, __HIP_MEMORY_SCOPE_AGENT);
    if (lane == 0)
        __hip_atomic_fetch_add(cnt + dst, 1.0f, __ATOMIC_RELAXED, __HIP_MEMORY_SCOPE_AGENT);
}

// ---------------- kernel: posterior + likelihood via WMMA f32 16x16x4
// block = 256 threads = 8 waves; wave g handles mixture-group g of a 16-node tile.
// D[node, i] = sum_l aggr[node, l] * Q[i, l, g]   (A = aggr tile, B = Q^T)
__global__ void __launch_bounds__(256)
k_posterior(const int* __restrict__ x,
            const float* __restrict__ summed,
            const float* __restrict__ cnt,
            const float* __restrict__ Qs,
            const float* __restrict__ Bs,
            float* __restrict__ out,
            int N) {
    __shared__ int sx[16];
    int base = blockIdx.x * 16;
    if (base >= N) return;
    if (threadIdx.x < 16) {
        int n = base + threadIdx.x;
        sx[threadIdx.x] = x[n < N ? n : (N - 1)];
    }
    __syncthreads();

    int lane = threadIdx.x & 31;
    int g    = threadIdx.x >> 5;   // wave id == mixture group
    int row  = lane & 15;
    int half = lane >> 4;

    int rm = base + row; if (rm >= N) rm = N - 1;            // clamped read row
    float inv = __builtin_amdgcn_rcpf(fmaxf(cnt[rm], 1.0f)); // scatter-mean divisor

    const float* srow = summed + (long)rm * ROW;        // aggr row (pre-division)
    const float* qrow = Qs + row * C_DIM * G_DIM;       // Q[i=row, :, :]

    v8f acc = {0.f, 0.f, 0.f, 0.f, 0.f, 0.f, 0.f, 0.f};
    #pragma unroll
    for (int kc = 0; kc < 4; ++kc) {
        int l0 = kc * 4 + half * 2;   // this lane's K-pair within the K=4 chunk
        v2f a, b;
        // A (16x4 f32): lane(row,half) holds A[row, l0], A[row, l0+1]
        a.x = srow[(l0 + 0) * G_DIM + g] * inv;
        a.y = srow[(l0 + 1) * G_DIM + g] * inv;
        // B (4x16 f32): lane(row,half) holds B[l0, i=row], B[l0+1, i=row] = Q[row, l, g]
        b.x = qrow[(l0 + 0) * G_DIM + g];
        b.y = qrow[(l0 + 1) * G_DIM + g];
        acc = __builtin_amdgcn_wmma_f32_16x16x4_f32(
                  /*neg_a=*/false, a, /*neg_b=*/false, b,
                  /*c_mod=*/(short)0, acc, /*reuse_a=*/false, /*reuse_b=*/false);
    }

    // C/D layout: vgpr r, lanes 0-15 -> M=r, lanes 16-31 -> M=r+8; N = lane%16.
    // Here M = node-in-tile, N = i (hidden state).
    float* post = out + (long)N * G_DIM;   // posterior block after likelihood [N,G]
    #pragma unroll
    for (int r = 0; r < 8; ++r) {
        int node = base + r + half * 8;
        int xm   = sx[r + half * 8];
        float bm = Bs[(row * M_DIM + xm) * G_DIM + g];   // B[i=row, x[node], g]
        float val = acc[r] * bm;                          // unnorm summed over l
        // sum over i (16 lanes of this half) with ds_swizzle XOR butterflies
        float tot = val;
        tot = SWZ_ADD(tot, 0x041F);   // xor 1
        tot = SWZ_ADD(tot, 0x081F);   // xor 2
        tot = SWZ_ADD(tot, 0x101F);   // xor 4
        tot = SWZ_ADD(tot, 0x201F);   // xor 8
        if (node < N) {
            post[((long)node * C_DIM + row) * G_DIM + g] =
                val * __builtin_amdgcn_rcpf(tot);
            if (row == 0)
                out[(long)node * G_DIM + g] = __logf(tot);
        }
    }
}

extern "C" void kernel_launch(void* const* d_in, const int* in_sizes, int n_in,
                              void* d_out, int out_size, void* d_ws, size_t ws_size,
                              hipStream_t stream) {
    const int*   x      = (const int*)d_in[0];
    const int*   ei     = (const int*)d_in[1];
    const float* prev_h = (const float*)d_in[2];
    const float* lamQ   = (const float*)d_in[3];
    const float* lamB   = (const float*)d_in[4];
    float* out = (float*)d_out;

    int N = in_sizes[0];
    int E = in_sizes[1] / 2;

    float* Qs     = (float*)d_ws;                    // 2048 floats
    float* Bs     = Qs + C_DIM * C_DIM * G_DIM;      // 4096 floats
    float* cnt    = Bs + C_DIM * M_DIM * G_DIM;      // N floats
    float* summed = cnt + N;                         // N*128 floats

    // zero cnt+summed (contiguous)
    long zn = (long)N * ROW + N;
    k_zero<<<(int)((zn + 255) / 256), 256, 0, stream>>>(cnt, zn);

    // tiny softmaxes
    k_softmax<<<1, 256, 0, stream>>>(lamQ, lamB, Qs, Bs);

    // edge scatter: 32 lanes per edge
    long st = (long)E * 32;
    k_scatter<<<(int)((st + 255) / 256), 256, 0, stream>>>(ei, prev_h, summed, cnt, E);

    // per-node posterior: 16 nodes per block, 8 waves (one per group)
    k_posterior<<<(N + 15) / 16, 256, 0, stream>>>(x, summed, cnt, Qs, Bs, out, N);
}